// FPModule_19207093748188
// MI455X (gfx1250) — compile-verified
//
#include <hip/hip_runtime.h>

typedef __attribute__((ext_vector_type(2))) float v2f;
typedef __attribute__((ext_vector_type(8))) float v8f;

#define N_PTS   8192
#define M_PTS   32768
#define C_IN    256
#define C_SKIP  64
#define K1_DIM  (C_IN + C_SKIP)   // 320
#define H_DIM   256
#define C_OUT   256

// ---------------------------------------------------------------------------
// Phase 1: brute-force 3-NN. One thread per query, pos staged through LDS.
// ---------------------------------------------------------------------------
#define KNN_BLOCK 256
#define KNN_TILE  1024

__global__ __launch_bounds__(KNN_BLOCK)
void knn_kernel(const float* __restrict__ pos,
                const float* __restrict__ pos_skip,
                int* __restrict__ knn_idx,
                float* __restrict__ knn_w) {
    __shared__ float sp[KNN_TILE * 3];
    const int m = blockIdx.x * KNN_BLOCK + threadIdx.x;   // M divisible by 256
    const float qx = pos_skip[m * 3 + 0];
    const float qy = pos_skip[m * 3 + 1];
    const float qz = pos_skip[m * 3 + 2];

    float b0 = 3.4e38f, b1 = 3.4e38f, b2 = 3.4e38f;
    int   i0 = 0, i1 = 0, i2 = 0;

    for (int t = 0; t < N_PTS; t += KNN_TILE) {
        for (int j = threadIdx.x; j < KNN_TILE * 3; j += KNN_BLOCK)
            sp[j] = pos[t * 3 + j];
        __syncthreads();
        #pragma unroll 4
        for (int j = 0; j < KNN_TILE; ++j) {
            const float dx = qx - sp[j * 3 + 0];
            const float dy = qy - sp[j * 3 + 1];
            const float dz = qz - sp[j * 3 + 2];
            const float d  = dx * dx + dy * dy + dz * dz;
            if (d < b2) {
                const int gi = t + j;
                if (d < b1) {
                    b2 = b1; i2 = i1;
                    if (d < b0) { b1 = b0; i1 = i0; b0 = d; i0 = gi; }
                    else        { b1 = d;  i1 = gi; }
                } else          { b2 = d;  i2 = gi; }
            }
        }
        __syncthreads();
    }

    knn_idx[m * 3 + 0] = i0;
    knn_idx[m * 3 + 1] = i1;
    knn_idx[m * 3 + 2] = i2;
    knn_w[m * 3 + 0] = 1.0f / fmaxf(b0, 1e-16f);
    knn_w[m * 3 + 1] = 1.0f / fmaxf(b1, 1e-16f);
    knn_w[m * 3 + 2] = 1.0f / fmaxf(b2, 1e-16f);
}

// ---------------------------------------------------------------------------
// Phase 2: gather + inverse-distance interpolation + concat with x_skip.
// ---------------------------------------------------------------------------
__global__ __launch_bounds__(C_IN)
void interp_concat_kernel(const float* __restrict__ x,
                          const float* __restrict__ x_skip,
                          const int* __restrict__ knn_idx,
                          const float* __restrict__ knn_w,
                          float* __restrict__ h_in) {
    const int m = blockIdx.x;
    const int c = threadIdx.x;
    __shared__ int   si[3];
    __shared__ float sw[3];
    if (c < 3) {
        si[c] = knn_idx[m * 3 + c];
        sw[c] = knn_w[m * 3 + c];
    }
    __syncthreads();
    const float wsum = sw[0] + sw[1] + sw[2];
    const float v = (sw[0] * x[si[0] * C_IN + c] +
                     sw[1] * x[si[1] * C_IN + c] +
                     sw[2] * x[si[2] * C_IN + c]) / wsum;
    h_in[m * K1_DIM + c] = v;
    if (c < C_SKIP)
        h_in[m * K1_DIM + C_IN + c] = x_skip[m * C_SKIP + c];
}

// ---------------------------------------------------------------------------
// Phase 3: fp32 WMMA GEMM  C[M,N] = A[M,K] * B[K,N] + bias (, ReLU)
//
// Block = 256 threads = 8 waves. The block stages the full B panel
// (KDIM x 64 columns) into LDS once, pair-major so a lane's B fragment
// (B[kk][col], B[kk+1][col]) is one ds_load_b64. Staging uses the gfx1250
// async Global->LDS path (ASYNCcnt) when available. Each wave computes a
// 32x64 output tile (8 f32 accumulators); per k-step:
// 2 global b64 (A) + 2 ds_load_2addr_b64 (B) -> 8 WMMAs.
// ---------------------------------------------------------------------------
template<int KDIM, bool RELU>
__global__ __launch_bounds__(256)
void gemm_wmma_kernel(const float* __restrict__ A,
                      const float* __restrict__ B,
                      const float* __restrict__ bias,
                      float* __restrict__ C,
                      int Ndim) {
    extern __shared__ float sbp[];     // KDIM/2 pairs * 64 cols * 2 floats

    const int tid   = threadIdx.x;
    const int lane  = tid & 31;
    const int wave  = tid >> 5;
    const int hlf   = lane >> 4;       // 0: lanes 0-15, 1: lanes 16-31
    const int l16   = lane & 15;

    const int nTiles64 = Ndim / 64;
    const int tileN    = blockIdx.x % nTiles64;
    const int mBlock   = blockIdx.x / nTiles64;      // 256-row slab
    const int rowBase  = mBlock * 256 + wave * 32;   // rows [rowBase, rowBase+32)
    const int colBase  = tileN * 64;

    // ---- stage B panel into LDS (pair-major) ----
    // sbp[(k>>1)*128 + c*2 + (k&1)] = B[k][colBase + c]
#if defined(__HIP_DEVICE_COMPILE__) && \
    __has_builtin(__builtin_amdgcn_global_load_async_to_lds_b32)
    for (int idx = tid; idx < KDIM * 64; idx += 256) {
        const int k = idx >> 6;         // 0..KDIM-1
        const int c = idx & 63;         // 0..63
        const float* gsrc = B + (size_t)k * Ndim + colBase + c;
        float*       ldst = &sbp[(k >> 1) * 128 + c * 2 + (k & 1)];
        const unsigned int ldsOff = (unsigned int)(uintptr_t)ldst; // AS3 offset
        __builtin_amdgcn_global_load_async_to_lds_b32(
            (__attribute__((address_space(1))) int*)(uintptr_t)gsrc,
            (__attribute__((address_space(3))) int*)ldsOff,
            0, 0);
    }
  #if __has_builtin(__builtin_amdgcn_s_wait_asynccnt)
    __builtin_amdgcn_s_wait_asynccnt(0);
  #else
    asm volatile("s_wait_asynccnt 0" ::: "memory");
  #endif
#else
    for (int idx = tid; idx < KDIM * 64; idx += 256) {
        const int k = idx >> 6;
        const int c = idx & 63;
        sbp[(k >> 1) * 128 + c * 2 + (k & 1)] = B[(size_t)k * Ndim + colBase + c];
    }
#endif
    __syncthreads();

    // A fragments: lane holds A[row][k + 2*hlf + {0,1}] (contiguous float2)
    const float* aRow0 = A + (size_t)(rowBase + l16) * KDIM + 2 * hlf;
    const float* aRow1 = aRow0 + (size_t)16 * KDIM;

    v8f acc00 = {}, acc01 = {}, acc02 = {}, acc03 = {};   // rows rowBase..+15
    v8f acc10 = {}, acc11 = {}, acc12 = {}, acc13 = {};   // rows rowBase+16..+31

    for (int k = 0; k < KDIM; k += 4) {
        const v2f a0 = *(const v2f*)(aRow0 + k);
        const v2f a1 = *(const v2f*)(aRow1 + k);
        // pair row for this lane: (k + 2*hlf)/2
        const float* bp = sbp + ((k >> 1) + hlf) * 128 + l16 * 2;

        // Load all four B fragments into distinct registers first so the
        // scheduler can issue both ds_load_2addr_b64 then 8 WMMAs in a row.
        const v2f b0 = *(const v2f*)(bp + 0);
        const v2f b1 = *(const v2f*)(bp + 32);
        const v2f b2 = *(const v2f*)(bp + 64);
        const v2f b3 = *(const v2f*)(bp + 96);

        acc00 = __builtin_amdgcn_wmma_f32_16x16x4_f32(false, a0, false, b0,
                                                      (short)0, acc00, false, false);
        acc10 = __builtin_amdgcn_wmma_f32_16x16x4_f32(false, a1, false, b0,
                                                      (short)0, acc10, false, false);
        acc01 = __builtin_amdgcn_wmma_f32_16x16x4_f32(false, a0, false, b1,
                                                      (short)0, acc01, false, false);
        acc11 = __builtin_amdgcn_wmma_f32_16x16x4_f32(false, a1, false, b1,
                                                      (short)0, acc11, false, false);
        acc02 = __builtin_amdgcn_wmma_f32_16x16x4_f32(false, a0, false, b2,
                                                      (short)0, acc02, false, false);
        acc12 = __builtin_amdgcn_wmma_f32_16x16x4_f32(false, a1, false, b2,
                                                      (short)0, acc12, false, false);
        acc03 = __builtin_amdgcn_wmma_f32_16x16x4_f32(false, a0, false, b3,
                                                      (short)0, acc03, false, false);
        acc13 = __builtin_amdgcn_wmma_f32_16x16x4_f32(false, a1, false, b3,
                                                      (short)0, acc13, false, false);
    }

    // D layout: VGPR j -> row (+ 8*hlf + j), col = colBase + nt*16 + l16
    float* c0 = C + (size_t)(rowBase + 8 * hlf) * Ndim;        // sub-tile 0
    float* c1 = C + (size_t)(rowBase + 16 + 8 * hlf) * Ndim;   // sub-tile 1

#define STORE_ACC(ACC, CB, NT)                                      \
    {                                                               \
        const int   col = colBase + (NT) * 16 + l16;                \
        const float bv  = bias[col];                                \
        _Pragma("unroll")                                           \
        for (int j = 0; j < 8; ++j) {                               \
            float v = ACC[j] + bv;                                  \
            if (RELU) v = fmaxf(v, 0.0f);                           \
            (CB)[(size_t)j * Ndim + col] = v;                       \
        }                                                           \
    }

    STORE_ACC(acc00, c0, 0)
    STORE_ACC(acc01, c0, 1)
    STORE_ACC(acc02, c0, 2)
    STORE_ACC(acc03, c0, 3)
    STORE_ACC(acc10, c1, 0)
    STORE_ACC(acc11, c1, 1)
    STORE_ACC(acc12, c1, 2)
    STORE_ACC(acc13, c1, 3)
#undef STORE_ACC
}

// ---------------------------------------------------------------------------
// Phase 4: copy pos_skip into the output tail.
// ---------------------------------------------------------------------------
__global__ void copy_pos_kernel(const float* __restrict__ pos_skip,
                                float* __restrict__ out_pos) {
    const int i = blockIdx.x * blockDim.x + threadIdx.x;
    if (i < M_PTS * 3) out_pos[i] = pos_skip[i];
}

// ---------------------------------------------------------------------------
extern "C" void kernel_launch(void* const* d_in, const int* in_sizes, int n_in,
                              void* d_out, int out_size, void* d_ws, size_t ws_size,
                              hipStream_t stream) {
    (void)in_sizes; (void)n_in; (void)out_size; (void)ws_size;

    const float* x        = (const float*)d_in[0];   // [N, 256]
    const float* pos      = (const float*)d_in[1];   // [N, 3]
    const float* x_skip   = (const float*)d_in[2];   // [M, 64]
    const float* pos_skip = (const float*)d_in[3];   // [M, 3]
    const float* W1       = (const float*)d_in[4];   // [320, 256]
    const float* b1       = (const float*)d_in[5];   // [256]
    const float* W2       = (const float*)d_in[6];   // [256, 256]
    const float* b2       = (const float*)d_in[7];   // [256]

    float* out = (float*)d_out;                      // [M,256] then [M,3]

    // Workspace carve-up
    char*  ws      = (char*)d_ws;
    size_t off     = 0;
    int*   knn_idx = (int*)(ws + off);   off += (size_t)M_PTS * 3 * sizeof(int);
    float* knn_w   = (float*)(ws + off); off += (size_t)M_PTS * 3 * sizeof(float);
    float* h_in    = (float*)(ws + off); off += (size_t)M_PTS * K1_DIM * sizeof(float);
    float* h_mid   = (float*)(ws + off); off += (size_t)M_PTS * H_DIM * sizeof(float);

    // 1) KNN
    knn_kernel<<<M_PTS / KNN_BLOCK, KNN_BLOCK, 0, stream>>>(pos, pos_skip,
                                                            knn_idx, knn_w);
    // 2) interpolate + concat
    interp_concat_kernel<<<M_PTS, C_IN, 0, stream>>>(x, x_skip, knn_idx,
                                                     knn_w, h_in);
    // 3) GEMM1 + bias + ReLU : [M,320]x[320,256]
    {
        const int nBlocks = (M_PTS / 256) * (H_DIM / 64);        // 512
        const size_t ldsB = (size_t)K1_DIM * 64 * sizeof(float); // 80 KB
        gemm_wmma_kernel<K1_DIM, true><<<nBlocks, 256, ldsB, stream>>>(
            h_in, W1, b1, h_mid, H_DIM);
    }
    // 4) GEMM2 + bias : [M,256]x[256,256]
    {
        const int nBlocks = (M_PTS / 256) * (C_OUT / 64);        // 512
        const size_t ldsB = (size_t)H_DIM * 64 * sizeof(float);  // 64 KB
        gemm_wmma_kernel<H_DIM, false><<<nBlocks, 256, ldsB, stream>>>(
            h_mid, W2, b2, out, C_OUT);
    }
    // 5) pos_skip passthrough
    copy_pos_kernel<<<(M_PTS * 3 + 255) / 256, 256, 0, stream>>>(
        pos_skip, out + (size_t)M_PTS * C_OUT);
}